// SLayer_2362232012851
// MI455X (gfx1250) — compile-verified
//
#include <hip/hip_runtime.h>

typedef float v2f __attribute__((ext_vector_type(2)));
typedef float v8f __attribute__((ext_vector_type(8)));

#define NB 64
#define NP 2048
#define NC 128
// D == 2 fixed

#define L2E 1.4426950408889634f   // log2(e): fold exp(x)=exp2(L2E*x) into WMMA weights

// One block: (batch b, 16-center tile nt). 256 threads = 8 waves.
// Each wave handles 256 points (16 WMMA p-tiles of 16 points).
// WMMA computes  log2e * (-dist)  =  A(centers,16x4) x B(point features,4x16) + C(bias),
// then v_exp_f32 (exp2) + mask-weighted accumulate.
__global__ __launch_bounds__(256) void slayer_wmma_kernel(
    const float* __restrict__ diagrams,   // [B,P,2]
    const float* __restrict__ masks,      // [B,P]
    const float* __restrict__ centers,    // [N,2]
    const float* __restrict__ sharp,      // [N,2]
    float* __restrict__ out)              // [B,N]
{
    const int nt   = blockIdx.x;          // 0..7 center tile
    const int b    = blockIdx.y;          // 0..63 batch
    const int tid  = threadIdx.x;         // 0..255
    const int lane = tid & 31;
    const int wave = tid >> 5;
    const int col  = lane & 15;           // A: center row m; B/C/D: point column n
    const bool hi  = (lane >= 16);
    const int moff = hi ? 8 : 0;          // C/D row offset for lanes 16..31

    __shared__ float abias[16];           // per-center bias  -L2E*(s0*c0^2 + s1*c1^2)
    __shared__ float red[16];             // per-center reduction

    // ---- loop-invariant A operand (centers), pre-scaled by log2(e) --------
    const int nc  = nt * 16 + col;
    const float c0 = centers[nc * 2 + 0];
    const float c1 = centers[nc * 2 + 1];
    const float s0 = sharp[nc * 2 + 0];
    const float s1 = sharp[nc * 2 + 1];

    v2f avec;                              // 16x4 f32 A layout (K=0,1 | K=2,3)
    avec[0] = L2E * (hi ? (-s0) : (2.0f * s0 * c0));
    avec[1] = L2E * (hi ? (-s1) : (2.0f * s1 * c1));

    if (tid < 16) {
        abias[tid] = -L2E * (s0 * c0 * c0 + s1 * c1 * c1);   // col==tid here
        red[tid]   = 0.0f;
    }
    __syncthreads();

    v8f cvec;                              // bias broadcast along columns
#pragma unroll
    for (int j = 0; j < 8; ++j) cvec[j] = abias[moff + j];

    float acc[8];
#pragma unroll
    for (int j = 0; j < 8; ++j) acc[j] = 0.0f;

    const float2* __restrict__ dia2 =
        reinterpret_cast<const float2*>(diagrams) + (size_t)b * NP;  // [P] float2
    const float*  __restrict__ mskb = masks + (size_t)b * NP;

    const int pbase = wave * 256 + col;    // this lane's point within the batch

    // ---- main loop: 16 WMMA tiles per wave --------------------------------
#pragma unroll 2
    for (int t = 0; t < 16; ++t) {
        const int p = pbase + t * 16;
        const float2 xy = dia2[p];         // global_load_b64
        const float  mk = mskb[p];         // lane-local mask (column-constant)

        v2f bvec;                          // 4x16 f32 B layout (K=0,1 | K=2,3)
        bvec[0] = hi ? (xy.x * xy.x) : xy.x;
        bvec[1] = hi ? (xy.y * xy.y) : xy.y;

        // D = A*B + C  ==  log2e * (-dist)
        v8f d = __builtin_amdgcn_wmma_f32_16x16x4_f32(
            /*neg_a=*/false, avec, /*neg_b=*/false, bvec,
            /*c_mod=*/(short)0, cvec, /*reuse_a=*/false, /*reuse_b=*/false);

#pragma unroll
        for (int j = 0; j < 8; ++j)
            acc[j] += __builtin_amdgcn_exp2f(d[j]) * mk;   // raw v_exp_f32
    }

    // ---- reduce across lanes & waves via LDS atomics ----------------------
#pragma unroll
    for (int j = 0; j < 8; ++j)
        atomicAdd(&red[moff + j], acc[j]);   // ds_add_f32 (no return)
    __syncthreads();

    if (tid < 16)
        out[(size_t)b * NC + nt * 16 + tid] = red[tid];
}

extern "C" void kernel_launch(void* const* d_in, const int* in_sizes, int n_in,
                              void* d_out, int out_size, void* d_ws, size_t ws_size,
                              hipStream_t stream) {
    const float* diagrams = (const float*)d_in[0];  // [64,2048,2]
    const float* masks    = (const float*)d_in[1];  // [64,2048]
    const float* centers  = (const float*)d_in[2];  // [128,2]
    const float* sharp    = (const float*)d_in[3];  // [128,2]
    float* out = (float*)d_out;                     // [64,128]

    dim3 grid(NC / 16, NB);   // (8 center tiles, 64 batches)
    dim3 block(256);          // 8 wave32
    slayer_wmma_kernel<<<grid, block, 0, stream>>>(diagrams, masks, centers, sharp, out);
}